// RoPE_1666447310943
// MI455X (gfx1250) — compile-verified
//
#include <hip/hip_runtime.h>
#include <math.h>

// RoPE for x:(B=2, H=16, SEQ=4096, D=128) fp32.
// Memory-bound: 128 MiB traffic @ 23.3 TB/s => ~5.5 us floor.
//
// Design (MI455X):
//  * One wave32 per (position, bh-chunk). Lane L owns the float4 at d=4L..4L+3,
//    i.e. rotation pairs with frequency indices 2L and 2L+1.
//  * sincos computed ONCE per lane (2 calls), reused across 8 batch*head
//    slices; trig latency overlapped with the inbound transfers.
//  * Preferred path: 8x global_load_async_to_lds_b128 (ASYNCcnt-tracked, no
//    dest VGPRs) -> staged s_wait_asynccnt drain -> ds_load_b128 -> rotate ->
//    non-temporal global_store_b128. Single-use streaming data is marked NT
//    so 128 MiB doesn't rinse through L2 for zero reuse.
//  * Fallback path (if async builtins absent): global_prefetch_b8 + NT b128
//    register pipeline.

typedef float v4f __attribute__((ext_vector_type(4)));
typedef int   v4i __attribute__((vector_size(16)));          // builtin's operand type
typedef __attribute__((address_space(1))) v4i* gptr_v4i;     // global
typedef __attribute__((address_space(3))) v4i* lptr_v4i;     // LDS

#define DK 128            // head dim
#define BH_PER_WAVE 8     // batch*head slices per wave

#if defined(__HIP_DEVICE_COMPILE__) && \
    __has_builtin(__builtin_amdgcn_global_load_async_to_lds_b128) && \
    __has_builtin(__builtin_amdgcn_s_wait_asynccnt)
#define USE_ASYNC_LDS 1
#else
#define USE_ASYNC_LDS 0
#endif

__global__ __launch_bounds__(256) void rope_f32_kernel(
    const float* __restrict__ x,
    const int*   __restrict__ tok,
    float*       __restrict__ out,
    int seq)
{
    const int lane      = threadIdx.x & 31;   // wave32
    const int waveInBlk = threadIdx.x >> 5;   // 8 waves / block
    const int pos       = blockIdx.x * 8 + waveInBlk;
    const int chunk     = blockIdx.y;         // which group of 8 bh slices
    if (pos >= seq) return;

    const v4f* __restrict__ x4 = (const v4f*)x;
    v4f*       __restrict__ o4 = (v4f*)out;

    const int rowF4 = pos * (DK / 4) + lane;              // 32 float4 per row
    const long long bhStrideF4 = (long long)seq * (DK / 4);
    const int bh0 = chunk * BH_PER_WAVE;

    long long idx[BH_PER_WAVE];
#pragma unroll
    for (int j = 0; j < BH_PER_WAVE; ++j)
        idx[j] = (long long)(bh0 + j) * bhStrideF4 + rowF4;

#if USE_ASYNC_LDS
    // 8 waves * 8 tiles * 32 lanes * 16B = 32 KB LDS (of 320 KB/WGP)
    __shared__ v4f smem[8 * BH_PER_WAVE * 32];
    const int tbase = waveInBlk * (BH_PER_WAVE * 32) + lane;
#pragma unroll
    for (int j = 0; j < BH_PER_WAVE; ++j) {
        // HBM -> LDS, ASYNCcnt-tracked, no destination VGPRs.
        __builtin_amdgcn_global_load_async_to_lds_b128(
            (gptr_v4i)(&x4[idx[j]]),
            (lptr_v4i)(&smem[tbase + j * 32]),
            /*offset=*/0, /*cpol=*/0);
    }
#else
#pragma unroll
    for (int j = 0; j < BH_PER_WAVE; ++j)
        __builtin_prefetch(&x4[idx[j]], 0, 3);            // global_prefetch_b8
#endif

    // ---- angles for this lane's two rotation pairs (overlaps transfers) ----
    const float p = (float)tok[pos];
    const float LOG2_THETA = 13.287712379549449f;         // log2(10000)
    const float e0 = exp2f(-((float)(2 * lane)     * (1.0f / 128.0f)) * LOG2_THETA);
    const float e1 = exp2f(-((float)(2 * lane + 1) * (1.0f / 128.0f)) * LOG2_THETA);
    float s0, c0, s1, c1;
    sincosf(p * e0, &s0, &c0);
    sincosf(p * e1, &s1, &c1);

    auto rotate_store = [&](v4f v, long long o) {
        v4f r;
        r.x = fmaf(c0, v.x, -s0 * v.y);                   // even = c*xe - s*xo
        r.y = fmaf(s0, v.x,  c0 * v.y);                   // odd  = s*xe + c*xo
        r.z = fmaf(c1, v.z, -s1 * v.w);
        r.w = fmaf(s1, v.z,  c1 * v.w);
        __builtin_nontemporal_store(r, &o4[o]);           // b128 store, th:NT
    };

#if USE_ASYNC_LDS
    // Async "done" returns in order: after asynccnt<=4, tiles 0..3 are in LDS.
    __builtin_amdgcn_s_wait_asynccnt(4);
#pragma unroll
    for (int j = 0; j < 4; ++j)
        rotate_store(smem[tbase + j * 32], idx[j]);       // ds_load_b128
    __builtin_amdgcn_s_wait_asynccnt(0);
#pragma unroll
    for (int j = 4; j < BH_PER_WAVE; ++j)
        rotate_store(smem[tbase + j * 32], idx[j]);
#else
    v4f v[BH_PER_WAVE];
#pragma unroll
    for (int j = 0; j < BH_PER_WAVE; ++j)
        v[j] = __builtin_nontemporal_load(&x4[idx[j]]);   // b128 load, th:NT
#pragma unroll
    for (int j = 0; j < BH_PER_WAVE; ++j)
        rotate_store(v[j], idx[j]);
#endif
}

extern "C" void kernel_launch(void* const* d_in, const int* in_sizes, int n_in,
                              void* d_out, int out_size, void* d_ws, size_t ws_size,
                              hipStream_t stream) {
    const float* x   = (const float*)d_in[0];
    const int*   tok = (const int*)d_in[1];
    float*       out = (float*)d_out;

    const int n   = in_sizes[0];          // 2*16*4096*128
    const int seq = in_sizes[1];          // 4096
    const int nbh = n / (seq * DK);       // 32 batch*head slices
    const int chunks = nbh / BH_PER_WAVE; // 4

    dim3 grid((seq + 7) / 8, chunks);     // 8 waves (positions) per 256-thread block
    rope_f32_kernel<<<grid, 256, 0, stream>>>(x, tok, out, seq);
}